// WassersteinLoss_68496138437002
// MI455X (gfx1250) — compile-verified
//
#include <hip/hip_runtime.h>
#include <math.h>

typedef __attribute__((ext_vector_type(2))) float v2f;
typedef __attribute__((ext_vector_type(4))) float v4f;
typedef __attribute__((ext_vector_type(8))) float v8f;

#define BSZ 128
#define NKB 128        // K-split: one block per chunk
#define KSTEP 64       // LDS staging depth
#define LSTR 70        // padded LDS row stride: (6*row+col)%64 injective -> conflict-free b64 frag loads
#define REG_EPS 0.01f
#define NUM_ITER 100

// ---------------- Kernel 1: K-split GEMM partials via V_WMMA_F32_16X16X4_F32 ----------------
__global__ __launch_bounds__(256) void wloss_gemm_partial(
    const float* __restrict__ imgs, const float* __restrict__ imgs_w,
    const float* __restrict__ target,
    float* __restrict__ xy_part, float* __restrict__ nx_part, float* __restrict__ ny_part,
    int K, int KC)
{
    __shared__ float sA[BSZ * LSTR];   // deltas tile, row-major [128][70]
    __shared__ float sB[BSZ * LSTR];   // target tile

    const int tid  = threadIdx.x;
    const int lane = tid & 31;
    const int wv   = tid >> 5;         // 8 waves
    const int m0   = wv * 16;          // wave's 16-row strip
    const int lm   = lane & 15;
    const int hb   = lane >> 4;        // lane half
    const int h2   = hb * 2;           // K sub-offset for A/B frags (K+0 / K+2)
    const int bk   = blockIdx.x;
    const long k0  = (long)bk * KC;

    v8f acc[8] = {};                   // 16x128 strip accumulator (64 VGPRs)
    float nX = 0.f, nY = 0.f;

    for (int kk0 = 0; kk0 < KC; kk0 += KSTEP) {
        __syncthreads();               // protect LDS reuse across chunks
        // ---- stage 128x64 of A=(imgs_w-imgs) and B=target into LDS (float4 global loads) ----
        #pragma unroll
        for (int i = 0; i < 8; ++i) {
            int idx = tid + i * 256;           // 2048 float4 slots
            int row = idx >> 4;
            int col = (idx & 15) * 4;
            long g  = (long)row * K + k0 + kk0 + col;
            v4f a  = *(const v4f*)(imgs_w + g);
            v4f b  = *(const v4f*)(imgs   + g);
            v4f t4 = *(const v4f*)(target + g);
            float* pa = &sA[row * LSTR + col];
            float* pb = &sB[row * LSTR + col];
            pa[0] = a.x - b.x; pa[1] = a.y - b.y; pa[2] = a.z - b.z; pa[3] = a.w - b.w;
            pb[0] = t4.x; pb[1] = t4.y; pb[2] = t4.z; pb[3] = t4.w;
        }
        __syncthreads();
        // ---- per-row squared-norm partials (wave-uniform branch, no intra-wave divergence) ----
        if (tid < 128) {
            const float* p = &sA[tid * LSTR];
            #pragma unroll 8
            for (int c = 0; c < KSTEP; ++c) { float v = p[c]; nX += v * v; }
        } else {
            const float* p = &sB[(tid - 128) * LSTR];
            #pragma unroll 8
            for (int c = 0; c < KSTEP; ++c) { float v = p[c]; nY += v * v; }
        }
        // ---- WMMA sweep: D += A(16x4) x B(4x16) over 8 N-tiles, K advances by 4 ----
        for (int kk = 0; kk < KSTEP; kk += 4) {
            v2f a = *(const v2f*)&sA[(m0 + lm) * LSTR + kk + h2];   // A frag: {K, K+1} per lane
            #pragma unroll
            for (int t = 0; t < 8; ++t) {
                v2f b = *(const v2f*)&sB[(t * 16 + lm) * LSTR + kk + h2]; // B(=Y^T) frag
                acc[t] = __builtin_amdgcn_wmma_f32_16x16x4_f32(
                    false, a, false, b, (short)0, acc[t], false, false);
            }
        }
    }

    // ---- write xy partial for this K-chunk (C/D layout: VGPR v -> M=m0+8*hb+v, N=16t+lm) ----
    float* dst = xy_part + (size_t)bk * (BSZ * BSZ);
    #pragma unroll
    for (int t = 0; t < 8; ++t) {
        #pragma unroll
        for (int v = 0; v < 8; ++v) {
            int M = m0 + hb * 8 + v;
            int N = t * 16 + lm;
            dst[M * BSZ + N] = acc[t][v];
        }
    }
    if (tid < 128) nx_part[bk * BSZ + tid] = nX;
    else           ny_part[bk * BSZ + (tid - 128)] = nY;
}

// ---------------- Kernel 2a: deterministic norm reduction ----------------
__global__ __launch_bounds__(256) void wloss_reduce_norms(
    const float* __restrict__ nx_part, const float* __restrict__ ny_part,
    float* __restrict__ x2, float* __restrict__ y2)
{
    int t = threadIdx.x;
    float s = 0.f;
    if (t < 128) {
        for (int b = 0; b < NKB; ++b) s += nx_part[b * BSZ + t];
        x2[t] = s;
    } else {
        int j = t - 128;
        for (int b = 0; b < NKB; ++b) s += ny_part[b * BSZ + j];
        y2[j] = s;
    }
}

// ---------------- Kernel 2b: reduce xy partials, form C = sqrt(max(x2+y2-2xy,0)) ----------------
__global__ __launch_bounds__(256) void wloss_combine(
    const float* __restrict__ xy_part, const float* __restrict__ x2,
    const float* __restrict__ y2, float* __restrict__ C)
{
    int e = blockIdx.x * 256 + threadIdx.x;   // 0..16383
    float s = 0.f;
    for (int b = 0; b < NKB; ++b) s += xy_part[(size_t)b * (BSZ * BSZ) + e];
    int i = e >> 7, j = e & 127;
    float d = x2[i] + y2[j] - 2.0f * s;
    C[e] = sqrtf(fmaxf(d, 0.0f));
}

// ---------------- Kernel 3: 100 Sinkhorn iterations, one workgroup, C in LDS ----------------
__global__ __launch_bounds__(128) void wloss_sinkhorn(
    const float* __restrict__ Cg, float* __restrict__ out)
{
    __shared__ float Cs[128 * 129];   // padded to kill column-phase bank conflicts
    __shared__ float us[128];
    __shared__ float vs[128];
    __shared__ float red[128];

    int t = threadIdx.x;
    for (int j = 0; j < 128; ++j) Cs[t * 129 + j] = Cg[t * 128 + j];
    us[t] = 0.f; vs[t] = 0.f;
    float ui = 0.f, vj = 0.f;
    const float log_ab = -logf(128.0f);   // log(1/128)
    __syncthreads();

    for (int it = 0; it < NUM_ITER; ++it) {
        // u = REG*(log_a - LSE_j(C - u - v))   [old v]
        {
            float m = -3.402823466e38f;
            for (int j = 0; j < 128; ++j) {
                float x = Cs[t * 129 + j] - ui - vs[j];
                m = fmaxf(m, x);
            }
            float s = 0.f;
            for (int j = 0; j < 128; ++j) {
                float x = Cs[t * 129 + j] - ui - vs[j];
                s += expf(x - m);
            }
            ui = REG_EPS * (log_ab - (m + logf(s)));
        }
        us[t] = ui;
        __syncthreads();
        // v = REG*(log_b - LSE_i(C - u_new - v))  [new u, old v]
        {
            float m = -3.402823466e38f;
            for (int i = 0; i < 128; ++i) {
                float x = Cs[i * 129 + t] - us[i] - vj;
                m = fmaxf(m, x);
            }
            float s = 0.f;
            for (int i = 0; i < 128; ++i) {
                float x = Cs[i * 129 + t] - us[i] - vj;
                s += expf(x - m);
            }
            vj = REG_EPS * (log_ab - (m + logf(s)));
        }
        vs[t] = vj;
        __syncthreads();
    }

    red[t] = (ui + vj) * (1.0f / 128.0f);   // sum(u*a + v*b), a=b=1/128
    __syncthreads();
    for (int off = 64; off > 0; off >>= 1) {
        if (t < off) red[t] += red[t + off];
        __syncthreads();
    }
    if (t == 0) out[0] = red[0];
}

extern "C" void kernel_launch(void* const* d_in, const int* in_sizes, int n_in,
                              void* d_out, int out_size, void* d_ws, size_t ws_size,
                              hipStream_t stream) {
    const float* imgs   = (const float*)d_in[0];
    const float* imgs_w = (const float*)d_in[1];
    const float* target = (const float*)d_in[2];
    int K  = in_sizes[0] / BSZ;   // 196608
    int KC = K / NKB;             // 1536 (24 chunks of 64)

    char* ws = (char*)d_ws;
    float* xy_part = (float*)ws; ws += (size_t)NKB * BSZ * BSZ * sizeof(float); // 8 MB
    float* nx      = (float*)ws; ws += (size_t)NKB * BSZ * sizeof(float);
    float* ny      = (float*)ws; ws += (size_t)NKB * BSZ * sizeof(float);
    float* x2      = (float*)ws; ws += BSZ * sizeof(float);
    float* y2      = (float*)ws; ws += BSZ * sizeof(float);
    float* C       = (float*)ws; ws += (size_t)BSZ * BSZ * sizeof(float);

    wloss_gemm_partial<<<NKB, 256, 0, stream>>>(imgs, imgs_w, target, xy_part, nx, ny, K, KC);
    wloss_reduce_norms<<<1, 256, 0, stream>>>(nx, ny, x2, y2);
    wloss_combine<<<64, 256, 0, stream>>>(xy_part, x2, y2, C);
    wloss_sinkhorn<<<1, 128, 0, stream>>>(C, (float*)d_out);
}